// DownBlock_25383256720205
// MI455X (gfx1250) — compile-verified
//
#include <hip/hip_runtime.h>

// ---------------------------------------------------------------------------
// MI455X / gfx1250 implementation of the icosahedral DownBlock:
//   xp = mean-gather(x, down_neigh)                    (B,32,40962)
//   h1 = lrelu(bn(conv7(xp, conv_neigh, W1)+b1))       (B,64,40962)
//   out= lrelu(bn(conv7(h1, conv_neigh, W2)+b2))       (B,64,40962)
// conv7 == GEMM  (M = B*V, N = 64, K = C*7) executed with
// V_WMMA_F32_16X16X4_F32 (full fp32 — problem is HBM/gather bound, ~56 GFLOP
// vs ~1.2 GB traffic, so fp32 matrix cores are already far from the limit).
// Staging copies that need no elementwise transform use gfx1250's async
// global->LDS path (GLOBAL_LOAD_ASYNC_TO_LDS_B32 + s_wait_asynccnt), keeping
// per-lane LDS addressing so the bank-conflict-free 226-float row pad is kept.
// ---------------------------------------------------------------------------

typedef float v2f __attribute__((ext_vector_type(2)));
typedef float v8f __attribute__((ext_vector_type(8)));

#define VC      40962      // coarse vertices
#define VF      163842     // fine vertices
#define NB      16         // batch
#define KN      7          // neighbors
#define C1      32         // input channels
#define OCH     64         // output channels
#define VTILE   64         // vertices per workgroup (4 waves x 16)
#define GPAD    226        // LDS row stride (224 + 2): even (8B-aligned b64
                           // loads) and 226%64=34 -> 16 lanes hit 16 banks
#define NVALID  (NB*VC)    // elements per channel for batch-norm stats

// ---- gfx1250 async global->LDS helpers ------------------------------------
// Per-lane: copy 4 bytes from the global address gptr to the LDS address lds_off; tracked by ASYNCcnt.
__device__ __forceinline__ void async_g2lds_b32(void* lds_ptr, const void* gptr) {
  unsigned lds_off = (unsigned)(size_t)lds_ptr;            // low 32 bits of the flat pointer are the LDS offset
  unsigned long long ga = (unsigned long long)(size_t)gptr;
  asm volatile("global_load_async_to_lds_b32 %0, %1, off"
               :: "v"(lds_off), "v"(ga) : "memory");
}
__device__ __forceinline__ void wait_async0() {
#if __has_builtin(__builtin_amdgcn_s_wait_asynccnt)
  __builtin_amdgcn_s_wait_asynccnt(0);
#else
  asm volatile("s_wait_asynccnt 0x0" ::: "memory");
#endif
}

// ------------------------------ utility kernels ----------------------------

__global__ void k_zero(float* __restrict__ p, int n) {
  int i = blockIdx.x * blockDim.x + threadIdx.x;
  if (i < n) p[i] = 0.f;
}

// xp[b,c,v] = mean_k x[b,c,dn[v,k]]
__global__ void k_down(const float* __restrict__ x, const int* __restrict__ dn,
                       float* __restrict__ xp) {
  long i = (long)blockIdx.x * blockDim.x + threadIdx.x;
  if (i >= (long)NB * C1 * VC) return;
  int  v  = (int)(i % VC);
  long bc = i / VC;
  const float* row = x + bc * (long)VF;
  const int*   nb  = dn + v * KN;
  float s = 0.f;
#pragma unroll
  for (int k = 0; k < KN; ++k) s += row[nb[k]];
  xp[i] = s * (1.0f / 7.0f);
}

// per-channel BN affine from accumulated sum / sum-of-squares
__global__ void k_stats(const float* __restrict__ sum, const float* __restrict__ sq,
                        const float* __restrict__ gamma, const float* __restrict__ beta,
                        float* __restrict__ scale, float* __restrict__ shift) {
  int c = threadIdx.x;
  if (c < OCH) {
    const float inv  = 1.0f / (float)NVALID;
    float mean = sum[c] * inv;
    float var  = sq[c] * inv - mean * mean;
    float sc   = gamma[c] * rsqrtf(var + 1e-5f);
    scale[c] = sc;
    shift[c] = beta[c] - mean * sc;
  }
}

// in-place BN + LeakyReLU on the final output
__global__ void k_final(float* __restrict__ out, const float* __restrict__ scale,
                        const float* __restrict__ shift) {
  long i = (long)blockIdx.x * blockDim.x + threadIdx.x;
  if (i >= (long)NB * OCH * VC) return;
  int c = (int)((i / VC) % OCH);
  float v = fmaf(out[i], scale[c], shift[c]);
  out[i] = v >= 0.f ? v : 0.2f * v;
}

// ------------------------------ fused conv GEMM ----------------------------
// One workgroup: 4 waves, batch b = blockIdx.y, vertices [v0, v0+64).
// Gathered activations + weight K-chunk staged in LDS; each wave owns a
// 16(v) x 64(o) tile accumulated by v_wmma_f32_16x16x4_f32.
// ACT=true applies the previous layer's BN+LeakyReLU at gather time (VALU
// staging path); ACT=false stages the gather with async global->LDS copies.
template <bool ACT>
__global__ __launch_bounds__(128)
void k_conv(const float* __restrict__ src, int nchunk,            // Cin = 32*nchunk
            const float* __restrict__ W,  const float* __restrict__ bias,
            const float* __restrict__ scIn, const float* __restrict__ shIn,
            const int* __restrict__ cn,
            float* __restrict__ out,
            float* __restrict__ sumOut, float* __restrict__ sqOut) {
  __shared__ float g_lds[VTILE * GPAD];   // gathered A tile (one K-chunk)
  __shared__ float w_lds[OCH * GPAD];     // weight B tile  (one K-chunk)
  __shared__ int   cn_lds[VTILE * KN];
  __shared__ float r_sum[OCH];
  __shared__ float r_sq[OCH];

  const int tid  = threadIdx.x;
  const int lane = tid & 31;
  const int wid  = tid >> 5;
  const int half = lane >> 4;
  const int l16  = lane & 15;
  const int b    = blockIdx.y;
  const int v0   = blockIdx.x * VTILE;
  const int Cin  = nchunk * 32;

  // neighbor indices for this vertex tile (clamped for the ragged last tile)
  for (int i = tid; i < VTILE * KN; i += 128) {
    int m = i / KN;
    int v = v0 + m; if (v >= VC) v = VC - 1;
    cn_lds[i] = cn[v * KN + (i - m * KN)];
  }
  if (tid < OCH) { r_sum[tid] = 0.f; r_sq[tid] = 0.f; }

  v8f acc[4];
#pragma unroll
  for (int ot = 0; ot < 4; ++ot) acc[ot] = (v8f){};

  const float* sb = src + (long)b * Cin * VC;

  for (int ch = 0; ch < nchunk; ++ch) {
    __syncthreads();
    // stage weight K-chunk (pure copy -> async global->LDS):
    //   w_lds[n*GPAD + k] = W[n, ch*224 + k]
    for (int i = tid; i < OCH * 224; i += 128) {
      int n = i / 224, k = i - n * 224;
      async_g2lds_b32(&w_lds[n * GPAD + k],
                      &W[(long)n * (Cin * KN) + ch * 224 + k]);
    }
    // stage gathered activation K-chunk: g[m, k] = act(src[b, c, cn[v,kk]])
    for (int i = tid; i < VTILE * 224; i += 128) {
      int m = i / 224, k = i - m * 224;
      int cl = k / KN, kk = k - cl * KN;
      int c  = ch * 32 + cl;
      const float* gp = &sb[(long)c * VC + cn_lds[m * KN + kk]];
      if (ACT) {  // previous layer BN + LeakyReLU (commutes with the gather)
        float val = fmaf(*gp, scIn[c], shIn[c]);
        g_lds[m * GPAD + k] = val >= 0.f ? val : 0.2f * val;
      } else {    // raw gather -> direct async scatter into LDS
        async_g2lds_b32(&g_lds[m * GPAD + k], gp);
      }
    }
    wait_async0();           // all async global->LDS copies of this chunk done
    __syncthreads();

    // K-loop: 56 steps of K=4. A fragment (16x4 f32, ISA layout: lanes 0-15
    // hold K={0,1}, lanes 16-31 hold K={2,3}) = one ds_load_b64 per step.
    const float* ga = &g_lds[(wid * 16 + l16) * GPAD + 2 * half];
    const float* wb = &w_lds[l16 * GPAD + 2 * half];
#pragma unroll 4
    for (int k0 = 0; k0 < 224; k0 += 4) {
      v2f a = *(const v2f*)(ga + k0);
#pragma unroll
      for (int ot = 0; ot < 4; ++ot) {
        v2f bf = *(const v2f*)(wb + ot * 16 * GPAD + k0);
        acc[ot] = __builtin_amdgcn_wmma_f32_16x16x4_f32(
            false, a, false, bf, (short)0, acc[ot], false, false);
      }
    }
  }

  // bias + store + per-channel batch-norm statistics.
  // D layout: VGPR r -> row M = r (lanes 0-15) / r+8 (lanes 16-31), col N = l16.
  const bool full  = (v0 + VTILE) <= VC;
  float* ob        = out + (long)b * OCH * VC;
  const int vbase  = v0 + wid * 16 + 8 * half;
#pragma unroll
  for (int ot = 0; ot < 4; ++ot) {
    int n = ot * 16 + l16;
    float bv = bias[n];
    float s = 0.f, q = 0.f;
    float* op = ob + (long)n * VC + vbase;
    if (full) {
#pragma unroll
      for (int r = 0; r < 8; r += 2) {   // paired b64 stores (VC%4==2 => no b128)
        v2f pr;
        pr.x = acc[ot][r] + bv;
        pr.y = acc[ot][r + 1] + bv;
        *(v2f*)(op + r) = pr;
        s += pr.x + pr.y;
        q += pr.x * pr.x + pr.y * pr.y;
      }
    } else {
#pragma unroll
      for (int r = 0; r < 8; ++r) {
        if (vbase + r < VC) {
          float val = acc[ot][r] + bv;
          op[r] = val; s += val; q += val * val;
        }
      }
    }
    atomicAdd(&r_sum[n], s);   // ds_add_f32
    atomicAdd(&r_sq[n], q);
  }
  __syncthreads();
  if (tid < OCH) {
    atomicAdd(sumOut + tid, r_sum[tid]);
    atomicAdd(sqOut + tid, r_sq[tid]);
  }
}

// ------------------------------ launch -------------------------------------

extern "C" void kernel_launch(void* const* d_in, const int* in_sizes, int n_in,
                              void* d_out, int out_size, void* d_ws, size_t ws_size,
                              hipStream_t stream) {
  const float* x   = (const float*)d_in[0];
  const float* w1  = (const float*)d_in[1];
  const float* b1  = (const float*)d_in[2];
  const float* g1  = (const float*)d_in[3];
  const float* be1 = (const float*)d_in[4];
  const float* w2  = (const float*)d_in[5];
  const float* b2  = (const float*)d_in[6];
  const float* g2  = (const float*)d_in[7];
  const float* be2 = (const float*)d_in[8];
  const int*   cn  = (const int*)d_in[9];    // conv_neigh_indices
  const int*   dn  = (const int*)d_in[10];   // down_neigh_indices
  float* out = (float*)d_out;

  // workspace: xp (84MB) | h1 (168MB) | stats (512 floats)
  float* ws = (float*)d_ws;
  const long XPN = (long)NB * C1 * VC;
  const long H1N = (long)NB * OCH * VC;
  float* xp = ws;
  float* h1 = ws + XPN;
  float* st = ws + XPN + H1N;
  float *sum1 = st,       *sq1 = st + 64,  *sum2 = st + 128, *sq2 = st + 192;
  float *sc1  = st + 256, *sh1 = st + 320, *sc2  = st + 384, *sh2 = st + 448;

  k_zero<<<1, 512, 0, stream>>>(st, 512);
  k_down<<<(int)((XPN + 255) / 256), 256, 0, stream>>>(x, dn, xp);

  dim3 cgrid((VC + VTILE - 1) / VTILE, NB);   // 641 x 16
  k_conv<false><<<cgrid, 128, 0, stream>>>(xp, 1, w1, b1, nullptr, nullptr,
                                           cn, h1, sum1, sq1);
  k_stats<<<1, 64, 0, stream>>>(sum1, sq1, g1, be1, sc1, sh1);
  k_conv<true><<<cgrid, 128, 0, stream>>>(h1, 2, w2, b2, sc1, sh1,
                                          cn, out, sum2, sq2);
  k_stats<<<1, 64, 0, stream>>>(sum2, sq2, g2, be2, sc2, sh2);
  k_final<<<(int)((H1N + 255) / 256), 256, 0, stream>>>(out, sc2, sh2);
}